// PSGNLoss_78700980732421
// MI455X (gfx1250) — compile-verified
//
#include <hip/hip_runtime.h>

typedef float v2f __attribute__((ext_vector_type(2)));
typedef float v8f __attribute__((ext_vector_type(8)));

#define BVIEWS   24
#define NPTS     1024
#define NT       1024          // 32 waves per block (wave32)
#define SINK_IT  20
#define EPSV     0.05f
#define INVEPS   20.0f
#define LOGMU    (-6.93147180559945f)   // -log(1024)
#define W_OPP    0.55f

// d_ws float layout
#define WS_COLS  0    // sum over (bv, gt_j) of min_i C_ij   (chamfer term 1)
#define WS_ROWS  1    // sum over (bv, pc_i) of min_j C_ij   (chamfer term 2, *W_OPP)
#define WS_EMD   2    // sum over bv of sum(P*C)
#define WS_PCM   16   // 24*3 pc means
#define WS_GTM   96   // 24*3 gt means

__device__ __forceinline__ float block_sum(float v, float* red) {
  const int t = threadIdx.x;
  red[t] = v;
  __syncthreads();
  #pragma unroll
  for (int s = NT >> 1; s > 0; s >>= 1) {
    if (t < s) red[t] += red[t + s];
    __syncthreads();
  }
  float r = red[0];
  __syncthreads();
  return r;
}

__global__ void psgn_init(float* __restrict__ ws) {
  if (threadIdx.x < 4) ws[threadIdx.x] = 0.0f;
}

__global__ __launch_bounds__(NT) void psgn_main(
    const float* __restrict__ pav,   // [24,1024,3] points_all_views
    const float* __restrict__ prs,   // [8,1024,3]  points_resample
    float* __restrict__ ws) {
  // WMMA operand planes, K padded to 4: plane 0 = (x,y), plane 1 = (z,0)
  __shared__ v2f apc[2][NPTS];     // pc as A operand
  __shared__ v2f agt[2][NPTS];     // gt as B operand
  __shared__ v2f rpf[NPTS];        // per pc row:  (|pc_i|^2, f_i)
  __shared__ v2f cpg[NPTS];        // per gt col:  (|gt_j|^2, g_j)
  __shared__ float red[NT];

  const int bv = blockIdx.x;
  const int t  = threadIdx.x;

  // ---------------- Phase 0: load, means, center, norms ----------------
  const float* p = pav + ((size_t)bv * NPTS + t) * 3;
  const float* q = prs + ((size_t)(bv / 3) * NPTS + t) * 3;
  float px = p[0], py = p[1], pz = p[2];
  float qx = q[0], qy = q[1], qz = q[2];

  const float inv_n = 1.0f / (float)NPTS;
  float mpx = block_sum(px, red) * inv_n;
  float mpy = block_sum(py, red) * inv_n;
  float mpz = block_sum(pz, red) * inv_n;
  float mqx = block_sum(qx, red) * inv_n;
  float mqy = block_sum(qy, red) * inv_n;
  float mqz = block_sum(qz, red) * inv_n;

  if (t == 0) {
    ws[WS_PCM + bv*3 + 0] = mpx; ws[WS_PCM + bv*3 + 1] = mpy; ws[WS_PCM + bv*3 + 2] = mpz;
    ws[WS_GTM + bv*3 + 0] = mqx; ws[WS_GTM + bv*3 + 1] = mqy; ws[WS_GTM + bv*3 + 2] = mqz;
  }

  px -= mpx; py -= mpy; pz -= mpz;
  qx -= mqx; qy -= mqy; qz -= mqz;
  const float ni = px*px + py*py + pz*pz;
  const float nq = qx*qx + qy*qy + qz*qz;
  {
    v2f v01; v01.x = px; v01.y = py;  apc[0][t] = v01;
    v2f v23; v23.x = pz; v23.y = 0.f; apc[1][t] = v23;
    v2f w01; w01.x = qx; w01.y = qy;  agt[0][t] = w01;
    v2f w23; w23.x = qz; w23.y = 0.f; agt[1][t] = w23;
    v2f rp;  rp.x = ni;  rp.y = 0.f;  rpf[t] = rp;
    v2f cg;  cg.x = nq;  cg.y = 0.f;  cpg[t] = cg;
  }
  __syncthreads();

  // ---------------- Phase 1: chamfer (both directions, LDS sweep) ------
  float rmin = 3.4e38f, cmin = 3.4e38f;
  for (int j = 0; j < NPTS; ++j) {
    v2f g01 = agt[0][j]; v2f g23 = agt[1][j];
    float dr = ni + cpg[j].x - 2.0f*(px*g01.x + py*g01.y + pz*g23.x);
    rmin = fminf(rmin, fmaxf(dr, 0.0f));           // pc point t vs all gt
    v2f p01 = apc[0][j]; v2f p23 = apc[1][j];
    float dc = nq + rpf[j].x - 2.0f*(qx*p01.x + qy*p01.y + qz*p23.x);
    cmin = fminf(cmin, fmaxf(dc, 0.0f));           // gt point t vs all pc
  }
  float rtot = block_sum(rmin, red);
  float ctot = block_sum(cmin, red);
  if (t == 0) { atomicAdd(&ws[WS_ROWS], rtot); atomicAdd(&ws[WS_COLS], ctot); }

  // ---------------- Phase 2: Sinkhorn, WMMA tiles + two-phase LSE ------
  const int wv  = t >> 5;       // wave 0..31
  const int ln  = t & 31;
  const int l15 = ln & 15;
  const int hi  = ln >> 4;      // half-wave selects K-plane and row group

  for (int it = 0; it < SINK_IT; ++it) {
    // ---- row pass: f_i = eps*(logmu - LSE_j((g_j - C_ij)/eps)) ----
    for (int tmi = 0; tmi < 2; ++tmi) {
      const int tm    = wv + tmi * 32;         // row tile 0..63
      const int arow  = (tm << 4) + l15;
      const int rbase = (tm << 4) + (hi << 3);
      const v2f a = apc[hi][arow];             // one ds_load_b64
      float ppcr[8], mx[8];
      #pragma unroll
      for (int r = 0; r < 8; ++r) { ppcr[r] = rpf[rbase + r].x; mx[r] = -3.4e38f; }

      // sweep 1: per-lane max of (g_j - C_ij)/eps  (no exp, no branches)
      for (int tn = 0; tn < 64; ++tn) {
        const int col = (tn << 4) + l15;
        const v2f bb  = agt[hi][col];
        const v2f pgj = cpg[col];              // (.x=|gt|^2, .y=g)
        v8f cz = {};
        v8f d = __builtin_amdgcn_wmma_f32_16x16x4_f32(
            false, a, false, bb, (short)0, cz, false, false);
        #pragma unroll
        for (int r = 0; r < 8; ++r) {
          float C = fmaxf(ppcr[r] + pgj.x - 2.0f * d[r], 0.0f);
          mx[r] = fmaxf(mx[r], (pgj.y - C) * INVEPS);
        }
      }
      // full-row max across the 16 lanes of each half-wave
      #pragma unroll
      for (int r = 0; r < 8; ++r) {
        #pragma unroll
        for (int msk = 1; msk <= 8; msk <<= 1)
          mx[r] = fmaxf(mx[r], __shfl_xor(mx[r], msk, 32));
      }
      // sweep 2: sum exp(v - max)
      float sm[8];
      #pragma unroll
      for (int r = 0; r < 8; ++r) sm[r] = 0.0f;
      for (int tn = 0; tn < 64; ++tn) {
        const int col = (tn << 4) + l15;
        const v2f bb  = agt[hi][col];
        const v2f pgj = cpg[col];
        v8f cz = {};
        v8f d = __builtin_amdgcn_wmma_f32_16x16x4_f32(
            false, a, false, bb, (short)0, cz, false, false);
        #pragma unroll
        for (int r = 0; r < 8; ++r) {
          float C = fmaxf(ppcr[r] + pgj.x - 2.0f * d[r], 0.0f);
          sm[r] += __expf((pgj.y - C) * INVEPS - mx[r]);
        }
      }
      #pragma unroll
      for (int r = 0; r < 8; ++r) {
        #pragma unroll
        for (int msk = 1; msk <= 8; msk <<= 1)
          sm[r] += __shfl_xor(sm[r], msk, 32);
        if (l15 == 0)
          rpf[rbase + r].y = EPSV * (LOGMU - (mx[r] + __logf(sm[r])));
      }
    }
    __syncthreads();

    // ---- col pass: g_j = eps*(lognu - LSE_i((f_i - C_ij)/eps)) ----
    for (int tni = 0; tni < 2; ++tni) {
      const int tn  = wv + tni * 32;
      const int col = (tn << 4) + l15;
      const v2f bb  = agt[hi][col];
      const float pgc = cpg[col].x;
      float mx = -3.4e38f;

      for (int tm = 0; tm < 64; ++tm) {
        const v2f a = apc[hi][(tm << 4) + l15];
        v8f cz = {};
        v8f d = __builtin_amdgcn_wmma_f32_16x16x4_f32(
            false, a, false, bb, (short)0, cz, false, false);
        const int rbase = (tm << 4) + (hi << 3);
        #pragma unroll
        for (int r = 0; r < 8; ++r) {
          v2f rr = rpf[rbase + r];             // (.x=|pc|^2, .y=f)
          float C = fmaxf(rr.x + pgc - 2.0f * d[r], 0.0f);
          mx = fmaxf(mx, (rr.y - C) * INVEPS);
        }
      }
      mx = fmaxf(mx, __shfl_xor(mx, 16, 32));  // lanes L,L+16 share the column

      float sm = 0.0f;
      for (int tm = 0; tm < 64; ++tm) {
        const v2f a = apc[hi][(tm << 4) + l15];
        v8f cz = {};
        v8f d = __builtin_amdgcn_wmma_f32_16x16x4_f32(
            false, a, false, bb, (short)0, cz, false, false);
        const int rbase = (tm << 4) + (hi << 3);
        #pragma unroll
        for (int r = 0; r < 8; ++r) {
          v2f rr = rpf[rbase + r];
          float C = fmaxf(rr.x + pgc - 2.0f * d[r], 0.0f);
          sm += __expf((rr.y - C) * INVEPS - mx);
        }
      }
      sm += __shfl_xor(sm, 16, 32);
      if (hi == 0)
        cpg[col].y = EPSV * (LOGMU - (mx + __logf(sm)));
    }
    __syncthreads();
  }

  // ---------------- Phase 3: transport cost sum(P*C) -------------------
  float acc = 0.0f;
  for (int tmi = 0; tmi < 2; ++tmi) {
    const int tm    = wv + tmi * 32;
    const int arow  = (tm << 4) + l15;
    const int rbase = (tm << 4) + (hi << 3);
    const v2f a = apc[hi][arow];
    float ppcr[8], fr[8];
    #pragma unroll
    for (int r = 0; r < 8; ++r) {
      v2f rr = rpf[rbase + r];
      ppcr[r] = rr.x; fr[r] = rr.y;
    }
    for (int tn = 0; tn < 64; ++tn) {
      const int col = (tn << 4) + l15;
      const v2f bb  = agt[hi][col];
      const v2f pgj = cpg[col];
      v8f cz = {};
      v8f d = __builtin_amdgcn_wmma_f32_16x16x4_f32(
          false, a, false, bb, (short)0, cz, false, false);
      #pragma unroll
      for (int r = 0; r < 8; ++r) {
        float C = fmaxf(ppcr[r] + pgj.x - 2.0f * d[r], 0.0f);
        acc += __expf((fr[r] + pgj.y - C) * INVEPS) * C;
      }
    }
  }
  float etot = block_sum(acc, red);
  if (t == 0) atomicAdd(&ws[WS_EMD], etot);
}

__global__ void psgn_final(const float* __restrict__ ws, float* __restrict__ out) {
  const float inv_bn = 1.0f / (float)(BVIEWS * NPTS);
  float chamfer = ws[WS_COLS] * inv_bn + W_OPP * ws[WS_ROWS] * inv_bn;
  float emd     = ws[WS_EMD] * (1.0f / (float)BVIEWS);
  float ml = 0.0f;
  for (int i = 0; i < BVIEWS * 3; ++i) {
    float diff = fabsf(ws[WS_PCM + i] - ws[WS_GTM + i]);
    ml += (diff < 1.0f) ? 0.5f * diff * diff : (diff - 0.5f);
  }
  ml *= 1.0f / (float)(BVIEWS * 3);
  out[0] = chamfer + emd + ml;  // loss
  out[1] = chamfer;
  out[2] = emd;
  out[3] = ml;
}

extern "C" void kernel_launch(void* const* d_in, const int* in_sizes, int n_in,
                              void* d_out, int out_size, void* d_ws, size_t ws_size,
                              hipStream_t stream) {
  const float* pav = (const float*)d_in[0];   // [24,1024,3]
  const float* prs = (const float*)d_in[1];   // [8,1024,3]
  float* ws  = (float*)d_ws;
  float* out = (float*)d_out;

  psgn_init<<<1, 32, 0, stream>>>(ws);
  psgn_main<<<BVIEWS, NT, 0, stream>>>(pav, prs, ws);
  psgn_final<<<1, 1, 0, stream>>>(ws, out);
}